// SudokuIterate_55860344651781
// MI455X (gfx1250) — compile-verified
//
#include <hip/hip_runtime.h>

#define KDIM   9
#define CELLS  81
#define BOARD  729
#define WAVES_PER_BLOCK 8
#define THREADS (WAVES_PER_BLOCK * 32)

#if defined(__has_builtin)
#  if __has_builtin(__builtin_amdgcn_global_load_async_to_lds_b32)
#    define USE_ASYNC_LDS 1
#  endif
#endif

typedef __attribute__((address_space(1))) int gint_t;   // global int*
typedef __attribute__((address_space(3))) int lint_t;   // LDS int*

static __device__ __forceinline__ void wait_asynccnt0() {
#if defined(__has_builtin)
#  if __has_builtin(__builtin_amdgcn_s_wait_asynccnt)
    __builtin_amdgcn_s_wait_asynccnt(0);
    return;
#  endif
#endif
#ifdef USE_ASYNC_LDS
    asm volatile("s_wait_asynccnt 0" ::: "memory");
#endif
}

__global__ __launch_bounds__(THREADS)
void sudoku_iter_kernel(const float* __restrict__ sudoku,
                        const float* __restrict__ rmask_in,
                        const float* __restrict__ rindex_in,
                        const float* __restrict__ conv_w,
                        const float* __restrict__ conv_b,
                        float* __restrict__ out_sudoku,
                        float* __restrict__ out_rmask,
                        float* __restrict__ out_rindex,
                        int nboards)
{
    __shared__ float smem[WAVES_PER_BLOCK][BOARD];
    const int lane  = threadIdx.x & 31;
    const int wave  = threadIdx.x >> 5;
    const int board = blockIdx.x * WAVES_PER_BLOCK + wave;
    const bool active = (board < nboards);

    float* lds = smem[wave];
    const float* gb = sudoku + (size_t)board * BOARD;

    // ---- Phase 1: stage board into LDS (async global->LDS, ASYNCcnt-tracked) ----
    if (active) {
#ifdef USE_ASYNC_LDS
        #pragma unroll
        for (int t = 0; t < 23; ++t) {
            const int e = lane + 32 * t;
            if (e < BOARD) {
                __builtin_amdgcn_global_load_async_to_lds_b32(
                    (gint_t*)(gb + e), (lint_t*)(lds + e), 0, 0);
            }
        }
#else
        #pragma unroll
        for (int t = 0; t < 23; ++t) {
            const int e = lane + 32 * t;
            if (e < BOARD) lds[e] = gb[e];
        }
#endif
    }
    wait_asynccnt0();
    __syncthreads();

    // ---- Weights ----
    float w[KDIM];
    #pragma unroll
    for (int c = 0; c < KDIM; ++c) w[c] = conv_w[c];
    const float bias = conv_b[0];

    // ---- Phase 2: per-cell score nic, wave argmax over 81 cells ----
    float bval = -3.4e38f;
    int   bp   = CELLS;
    if (active) {
        #pragma unroll
        for (int t = 0; t < 3; ++t) {
            const int p = lane + 32 * t;
            if (p < CELLS) {
                float sum = bias;
                #pragma unroll
                for (int c = 0; c < KDIM; ++c) sum += w[c] * lds[c * CELLS + p];
                const float nic0 = fmaxf(sum - 1.0f, 0.0f);
                const float m0   = fminf(fmaxf(1.0f - fabsf(nic0), 0.0f), 1.0f) * (-(float)KDIM);
                const float nic  = m0 - nic0;
                // ascending p => strict '>' keeps first occurrence
                if (nic > bval) { bval = nic; bp = p; }
            }
        }
    }
    #pragma unroll
    for (int off = 16; off >= 1; off >>= 1) {
        const float oval = __shfl_xor(bval, off, 32);
        const int   op   = __shfl_xor(bp,   off, 32);
        if (oval > bval || (oval == bval && op < bp)) { bval = oval; bp = op; }
    }
    const int pstar = bp;

    // ---- Phase 3: channel argmax at chosen cell (LDS broadcast reads) ----
    float mval = -3.4e38f;
    float wm   = 0.0f;
    int   midx = 0;
    if (active) {
        #pragma unroll
        for (int c = 0; c < KDIM; ++c) {
            const float v = lds[c * CELLS + pstar];
            if (v > mval) { mval = v; midx = c; wm = w[c]; }  // first occurrence
        }
    }

    // cell_mask: chosen cell = clip(w[midx]*mval + bias), elsewhere = clip(bias)
    const float cmStar = fminf(fmaxf(wm * mval + bias, 0.0f), 1.0f);
    const float cmElse = fminf(fmaxf(bias, 0.0f), 1.0f);

    // ---- Phase 4: streaming elementwise outputs (no div; c is compile-time) ----
    if (active) {
        const float ri  = rindex_in[board];
        const float rim = ri - 1.0f;
        const float* rmb = rmask_in  + (size_t)board * BOARD;
        float* osb = out_sudoku + (size_t)board * BOARD;
        float* orb = out_rmask  + (size_t)board * BOARD;
        #pragma unroll
        for (int c = 0; c < KDIM; ++c) {
            const bool cIsM = (c == midx);
            #pragma unroll
            for (int t = 0; t < 3; ++t) {
                const int p = lane + 32 * t;
                if (p < CELLS) {   // only the t==2 step is actually guarded
                    const int e = c * CELLS + p;
                    const bool atCell = (p == pstar);
                    const float s  = lds[e];
                    const float ov = (atCell && cIsM) ? mval : 0.0f;
                    const float cm = atCell ? cmStar : cmElse;
                    const float orm = s * cm * (1.0f - ov);
                    float rm = __builtin_nontemporal_load(rmb + e);
                    rm = fmaxf(rm, ri * orm);
                    rm = fmaxf(rm, rim * ov);
                    __builtin_nontemporal_store(s * (1.0f - orm), osb + e);
                    __builtin_nontemporal_store(rm, orb + e);
                }
            }
        }
        if (lane == 0) out_rindex[board] = ri + 1.0f;
    }
}

extern "C" void kernel_launch(void* const* d_in, const int* in_sizes, int n_in,
                              void* d_out, int out_size, void* d_ws, size_t ws_size,
                              hipStream_t stream) {
    const float* sudoku = (const float*)d_in[0];
    const float* rmask  = (const float*)d_in[1];
    const float* rindex = (const float*)d_in[2];
    const float* conv_w = (const float*)d_in[3];
    const float* conv_b = (const float*)d_in[4];

    const int nboards = in_sizes[0] / BOARD;

    float* out   = (float*)d_out;
    float* out_s = out;
    float* out_m = out + (size_t)nboards * BOARD;
    float* out_i = out + 2 * (size_t)nboards * BOARD;

    const int blocks = (nboards + WAVES_PER_BLOCK - 1) / WAVES_PER_BLOCK;
    hipLaunchKernelGGL(sudoku_iter_kernel, dim3(blocks), dim3(THREADS), 0, stream,
                       sudoku, rmask, rindex, conv_w, conv_b,
                       out_s, out_m, out_i, nboards);
}